// TripletLoss_87883620811397
// MI455X (gfx1250) — compile-verified
//
#include <hip/hip_runtime.h>
#include <stdint.h>

#define NROWS  4096
#define KDIM   1024
#define ASTRIDE (KDIM + 4)   // padded LDS row stride (4 x 1-DWORD TDM pads per row)
#define MTILE  16
#define NWAVES 8
#define JB     4
#define NTILES (NROWS / MTILE)
#define MARGINF 0.3f
#define EPSF    1e-12f

typedef __attribute__((ext_vector_type(2))) float    v2f;
typedef __attribute__((ext_vector_type(8))) float    v8f;
typedef __attribute__((ext_vector_type(4))) uint32_t u32x4;
typedef __attribute__((ext_vector_type(4))) int      i32x4;
typedef __attribute__((ext_vector_type(8))) int      i32x8;

// ------- bijective per-pair uniform (murmur3 fmix32; tie-free per stream) ----
__device__ __forceinline__ uint32_t fmix32(uint32_t h) {
  h ^= h >> 16; h *= 0x85EBCA6Bu;
  h ^= h >> 13; h *= 0xC2B2AE35u;
  h ^= h >> 16;
  return h;
}
__device__ __forceinline__ float pair_u01(uint32_t i, uint32_t j, uint32_t stream) {
  uint32_t h = fmix32(((i << 12) | j) * 0x9E3779B9u + stream);
  return __uint_as_float(0x3F800000u | (h >> 9)) - 1.0f;  // [0,1)
}

// ---------------- row squared-norms ----------------
__global__ void row_sqnorm_kernel(const float* __restrict__ X,
                                  float* __restrict__ sq) {
  const int row  = blockIdx.x;
  const int lane = threadIdx.x;  // one wave32 per row
  const float* p = X + (size_t)row * KDIM;
  float s = 0.0f;
#pragma unroll 4
  for (int k = lane; k < KDIM; k += 32) {
    float x = p[k];
    s = fmaf(x, x, s);
  }
#pragma unroll
  for (int m = 16; m >= 1; m >>= 1) s += __shfl_xor(s, m, 32);
  if (lane == 0) sq[row] = s;
}

// ---- 16x16 Gram tiles vs 4 j-tiles; A from padded LDS; rolled pipelined loop.
// kd is a RUNTIME value so the trip count is opaque (no full unroll / VGPR blowup).
// A index uses k + (k>>8): one pad DWORD after each 256-DWORD TDM segment.
__device__ __forceinline__ void gemm_jb4(const float* __restrict__ sArow,
                                         const float* __restrict__ B0,
                                         const float* __restrict__ B1,
                                         const float* __restrict__ B2,
                                         const float* __restrict__ B3,
                                         int kd,
                                         v8f& c0, v8f& c1, v8f& c2, v8f& c3) {
  v2f a  = *(const v2f*)(sArow);
  v2f b0 = *(const v2f*)(B0);
  v2f b1 = *(const v2f*)(B1);
  v2f b2 = *(const v2f*)(B2);
  v2f b3 = *(const v2f*)(B3);
#pragma unroll 4
  for (int k = 4; k < kd; k += 4) {
    const int ka = k + (k >> 8);
    v2f an  = *(const v2f*)(sArow + ka);
    v2f bn0 = *(const v2f*)(B0 + k);
    v2f bn1 = *(const v2f*)(B1 + k);
    v2f bn2 = *(const v2f*)(B2 + k);
    v2f bn3 = *(const v2f*)(B3 + k);
    c0 = __builtin_amdgcn_wmma_f32_16x16x4_f32(false, a, false, b0, (short)0, c0, false, false);
    c1 = __builtin_amdgcn_wmma_f32_16x16x4_f32(false, a, false, b1, (short)0, c1, false, false);
    c2 = __builtin_amdgcn_wmma_f32_16x16x4_f32(false, a, false, b2, (short)0, c2, false, false);
    c3 = __builtin_amdgcn_wmma_f32_16x16x4_f32(false, a, false, b3, (short)0, c3, false, false);
    a = an; b0 = bn0; b1 = bn1; b2 = bn2; b3 = bn3;
  }
  c0 = __builtin_amdgcn_wmma_f32_16x16x4_f32(false, a, false, b0, (short)0, c0, false, false);
  c1 = __builtin_amdgcn_wmma_f32_16x16x4_f32(false, a, false, b1, (short)0, c1, false, false);
  c2 = __builtin_amdgcn_wmma_f32_16x16x4_f32(false, a, false, b2, (short)0, c2, false, false);
  c3 = __builtin_amdgcn_wmma_f32_16x16x4_f32(false, a, false, b3, (short)0, c3, false, false);
}

// ---------------- fused triplet-loss main kernel ----------------
__global__ __launch_bounds__(32 * NWAVES)
void triplet_loss_kernel(const float* __restrict__ X,
                         const int*   __restrict__ tgt,
                         const float* __restrict__ sq,
                         float*       __restrict__ out,
                         int kd) {
  extern __shared__ char smem[];
  float* sA   = (float*)smem;  // [MTILE*ASTRIDE] floats at dynamic-LDS base (offset 0)
  float* s_u0 = (float*)(smem + (size_t)MTILE * ASTRIDE * sizeof(float));
  float* s_d0 = s_u0 + NWAVES * MTILE;
  float* s_u1 = s_d0 + NWAVES * MTILE;
  float* s_d1 = s_u1 + NWAVES * MTILE;
  float* s_dist_ap = s_d1 + NWAVES * MTILE;
  float* s_rand_an = s_dist_ap + MTILE;

  const int tid  = threadIdx.x;
  const int lane = tid & 31;
  const int wave = tid >> 5;
  const int half = lane >> 4;
  const int l16  = lane & 15;
  const int i0   = blockIdx.x * MTILE;

  // ---- stage the 16x1024 f32 A slab into LDS via the Tensor Data Mover,
  //      with 1 pad DWORD per 256 DWORDs -> row stride 1028 -> the 32 lanes'
  //      b64 reads hit all 64 LDS banks exactly once (conflict-free). ----
#if __has_builtin(__builtin_amdgcn_tensor_load_to_lds) && __has_builtin(__builtin_amdgcn_s_wait_tensorcnt)
  if (wave == 0) {
    uint64_t ga = (uint64_t)(uintptr_t)(X + (size_t)i0 * KDIM);
    u32x4 g0;
    g0[0] = 1u;                                                  // count=1 (valid D#)
    g0[1] = 0u;                                                  // lds_addr = 0
    g0[2] = (uint32_t)ga;                                        // global_addr[31:0]
    g0[3] = (uint32_t)((ga >> 32) & 0x01FFFFFFu) | 0x80000000u;  // addr[56:32] | type=2
    i32x8 g1;
    g1[0] = (int)((2u << 16)        // data_size = 4B
                | (1u << 20)        // pad_enable
                | (7u << 22)        // pad_interval code 7 = 256 DWORDs
                | (0u << 25));      // pad_amount  code 0 = 1 DWORD
    g1[1] = (int)(((uint32_t)KDIM & 0xFFFFu) << 16);             // tensor_dim0 lo16 (bits 63:48)
    g1[2] = (int)((((uint32_t)KDIM >> 16) & 0xFFFFu) |
                  ((uint32_t)MTILE << 16));                      // dim0 hi | tensor_dim1 lo
    g1[3] = (int)((uint32_t)KDIM << 16);                         // dim1 hi(0) | tile_dim0=1024
    g1[4] = (int)MTILE;                                          // tile_dim1=16, tile_dim2=0
    g1[5] = (int)KDIM;                                           // tensor_dim0_stride lo32
    g1[6] = 0;
    g1[7] = 0;
    i32x4 gz = {0, 0, 0, 0};
#if __clang_major__ >= 23
    i32x8 gz8 = {0, 0, 0, 0, 0, 0, 0, 0};
    __builtin_amdgcn_tensor_load_to_lds(g0, g1, gz, gz, gz8, 0);
#else
    __builtin_amdgcn_tensor_load_to_lds(g0, g1, gz, gz, 0);
#endif
    __builtin_amdgcn_s_wait_tensorcnt(0);
  }
#else
  for (int w = tid; w < MTILE * KDIM; w += 32 * NWAVES) {
    int r = w >> 10, k = w & (KDIM - 1);
    sA[r * ASTRIDE + k + (k >> 8)] = X[(size_t)i0 * KDIM + w];
  }
#endif
  __syncthreads();

  // Per-lane row metadata (C layout: lane holds rows v + 8*half, v=0..7)
  float sqi[8];
  int   ti[8];
#pragma unroll
  for (int v = 0; v < 8; ++v) {
    int r  = i0 + v + 8 * half;
    sqi[v] = sq[r];
    ti[v]  = tgt[r];
  }

  // A layout (16x4 f32): lane row = l16, K offset = 2*half — read from padded LDS
  const float* sArow = sA + (size_t)l16 * ASTRIDE + 2 * half;

  const uint32_t SP = 0xA511E9B3u;  // pass-1 stream (pos/neg choice)
  const uint32_t SH = 0x63D83595u;  // pass-2 stream (hard negatives)

  // ---------------- pass 1: sampled positive + random negative ----------------
  float b_apu[8], b_apd[8], b_anu[8], b_and[8];
#pragma unroll
  for (int v = 0; v < 8; ++v) {
    b_apu[v] = -2.0f; b_apd[v] = 0.0f;
    b_anu[v] = -2.0f; b_and[v] = 0.0f;
  }

  for (int g = wave * JB; g < NTILES; g += NWAVES * JB) {
    const int j0 = g * MTILE;
    const float* Bb = X + (size_t)(j0 + l16) * KDIM + 2 * half;
    float sqj[JB]; int tj[JB];
#pragma unroll
    for (int t = 0; t < JB; ++t) {
      const int j = j0 + t * MTILE + l16;
      sqj[t] = sq[j];
      tj[t]  = tgt[j];
    }
    v8f c[JB];
    c[0] = (v8f){}; c[1] = (v8f){}; c[2] = (v8f){}; c[3] = (v8f){};
    gemm_jb4(sArow, Bb, Bb + 16 * KDIM, Bb + 32 * KDIM, Bb + 48 * KDIM, kd,
             c[0], c[1], c[2], c[3]);
#pragma unroll
    for (int t = 0; t < JB; ++t) {
      const int j = j0 + t * MTILE + l16;
#pragma unroll
      for (int v = 0; v < 8; ++v) {
        const int row = i0 + v + 8 * half;
        float d = __builtin_amdgcn_sqrtf(fmaxf(sqi[v] + sqj[t] - 2.0f * c[t][v], EPSF));
        float u = pair_u01((uint32_t)row, (uint32_t)j, SP);
        bool  pos = (tj[t] == ti[v]);
        float up  = pos ? u : -1.0f;   // each (i,j) feeds exactly one argmax
        float un  = pos ? -1.0f : u;
        bool t0 = up > b_apu[v];
        b_apd[v] = t0 ? d : b_apd[v];  b_apu[v] = t0 ? up : b_apu[v];
        bool t1 = un > b_anu[v];
        b_and[v] = t1 ? d : b_and[v];  b_anu[v] = t1 ? un : b_anu[v];
      }
    }
  }

  // reduce over the 16 N-lanes per half, stash per wave in LDS
#pragma unroll
  for (int v = 0; v < 8; ++v) {
    float u0 = b_apu[v], d0 = b_apd[v];
    float u1 = b_anu[v], d1 = b_and[v];
#pragma unroll
    for (int m = 8; m >= 1; m >>= 1) {
      float ou = __shfl_xor(u0, m, 32), od = __shfl_xor(d0, m, 32);
      if (ou > u0) { u0 = ou; d0 = od; }
      ou = __shfl_xor(u1, m, 32); od = __shfl_xor(d1, m, 32);
      if (ou > u1) { u1 = ou; d1 = od; }
    }
    if (l16 == 0) {
      int row = v + 8 * half;
      s_u0[wave * MTILE + row] = u0; s_d0[wave * MTILE + row] = d0;
      s_u1[wave * MTILE + row] = u1; s_d1[wave * MTILE + row] = d1;
    }
  }
  __syncthreads();

  if (tid < MTILE) {
    float bu = -3.0f, bd = 0.0f, cu = -3.0f, cd = 0.0f;
#pragma unroll
    for (int w = 0; w < NWAVES; ++w) {
      if (s_u0[w * MTILE + tid] > bu) { bu = s_u0[w * MTILE + tid]; bd = s_d0[w * MTILE + tid]; }
      if (s_u1[w * MTILE + tid] > cu) { cu = s_u1[w * MTILE + tid]; cd = s_d1[w * MTILE + tid]; }
    }
    s_dist_ap[tid] = bd;
    s_rand_an[tid] = cd;
  }
  __syncthreads();

  // ---------------- pass 2: hard negatives ----------------
  float dap[8];
#pragma unroll
  for (int v = 0; v < 8; ++v) dap[v] = s_dist_ap[v + 8 * half];

  float b_hu[8], b_hd[8];
#pragma unroll
  for (int v = 0; v < 8; ++v) { b_hu[v] = -2.0f; b_hd[v] = 0.0f; }

  for (int g = wave * JB; g < NTILES; g += NWAVES * JB) {
    const int j0 = g * MTILE;
    const float* Bb = X + (size_t)(j0 + l16) * KDIM + 2 * half;
    float sqj[JB]; int tj[JB];
#pragma unroll
    for (int t = 0; t < JB; ++t) {
      const int j = j0 + t * MTILE + l16;
      sqj[t] = sq[j];
      tj[t]  = tgt[j];
    }
    v8f c[JB];
    c[0] = (v8f){}; c[1] = (v8f){}; c[2] = (v8f){}; c[3] = (v8f){};
    gemm_jb4(sArow, Bb, Bb + 16 * KDIM, Bb + 32 * KDIM, Bb + 48 * KDIM, kd,
             c[0], c[1], c[2], c[3]);
#pragma unroll
    for (int t = 0; t < JB; ++t) {
      const int j = j0 + t * MTILE + l16;
#pragma unroll
      for (int v = 0; v < 8; ++v) {
        const int row = i0 + v + 8 * half;
        float d = __builtin_amdgcn_sqrtf(fmaxf(sqi[v] + sqj[t] - 2.0f * c[t][v], EPSF));
        bool hard = (tj[t] != ti[v]) && (d > dap[v]);
        float uh = hard ? pair_u01((uint32_t)row, (uint32_t)j, SH) : -1.0f;
        bool tk = uh > b_hu[v];
        b_hd[v] = tk ? d : b_hd[v];  b_hu[v] = tk ? uh : b_hu[v];
      }
    }
  }

#pragma unroll
  for (int v = 0; v < 8; ++v) {
    float u0 = b_hu[v], d0 = b_hd[v];
#pragma unroll
    for (int m = 8; m >= 1; m >>= 1) {
      float ou = __shfl_xor(u0, m, 32), od = __shfl_xor(d0, m, 32);
      if (ou > u0) { u0 = ou; d0 = od; }
    }
    if (l16 == 0) {
      int row = v + 8 * half;
      s_u0[wave * MTILE + row] = u0; s_d0[wave * MTILE + row] = d0;
    }
  }
  __syncthreads();

  if (tid < MTILE) {
    float hu = -3.0f, hd = 0.0f;
#pragma unroll
    for (int w = 0; w < NWAVES; ++w)
      if (s_u0[w * MTILE + tid] > hu) { hu = s_u0[w * MTILE + tid]; hd = s_d0[w * MTILE + tid]; }
    float dist_an = (hu >= 0.0f) ? hd : s_rand_an[tid];
    float loss = fmaxf(s_dist_ap[tid] - dist_an + MARGINF, 0.0f);
    atomicAdd(out, loss * (1.0f / (float)NROWS));
  }
}

extern "C" void kernel_launch(void* const* d_in, const int* in_sizes, int n_in,
                              void* d_out, int out_size, void* d_ws, size_t ws_size,
                              hipStream_t stream) {
  (void)in_sizes; (void)n_in; (void)out_size; (void)ws_size;
  const float* X   = (const float*)d_in[0];
  const int*   tgt = (const int*)d_in[1];
  float*       out = (float*)d_out;
  float*       sq  = (float*)d_ws;  // 4096 floats = 16 KB scratch

  hipMemsetAsync(out, 0, sizeof(float), stream);
  row_sqnorm_kernel<<<NROWS, 32, 0, stream>>>(X, sq);

  size_t shmem = (size_t)MTILE * ASTRIDE * sizeof(float)         // padded A slab
               + (size_t)(4 * NWAVES * MTILE + 2 * MTILE) * sizeof(float);
  triplet_loss_kernel<<<NROWS / MTILE, 32 * NWAVES, shmem, stream>>>(X, tgt, sq, out, KDIM);
}